// SNPReductionNetModel_80144089743468
// MI455X (gfx1250) — compile-verified
//
#include <hip/hip_runtime.h>
#include <math.h>

// ---------------------------------------------------------------------------
// SNPReductionNet on MI455X (gfx1250, wave32)
//
// Stage 1 (HBM-bound, ~410MB of x):  blocked 80-wide dot products, staged
//   through LDS with float4 coalesced loads; deterministic per-tile
//   (sum, sumsq) partials for the LayerNorm (no float atomics).
// Stage 2: LayerNorm (in-place), one workgroup per batch row.
// Stage 3/4: dense 5000->1024->256 via V_WMMA_F32_16X16X4_F32 (full fp32
//   precision, matrix pipe), 16x16 tile per wave, 32x64 per workgroup.
// Stage 5: 256->1 projection, trivial.
// ---------------------------------------------------------------------------

typedef __attribute__((ext_vector_type(2))) float v2f;
typedef __attribute__((ext_vector_type(8))) float v8f;

#define BATCH     256
#define NBLK      5000
#define BLK_ELEMS 80
#define IN_DIM    400000
#define TILE_J    64
#define TILE_ELEMS (TILE_J * BLK_ELEMS)   // 5120 floats per x-tile
#define NTILES    79                       // ceil(5000 / 64)
#define D1        1024
#define D2        256

// ---------------------------------------------------------------------------
// Kernel 1: gblocks[b, j] = sum_{k=0}^{79} x[b, 80j+k] * sv[80j+k]
// plus per-(row, tile) partial (sum, sumsq) for LayerNorm.
// Grid: (NTILES, BATCH), Block: 256 threads.
// ---------------------------------------------------------------------------
__global__ __launch_bounds__(256)
void k_sparse_blocks(const float* __restrict__ x,
                     const float* __restrict__ sv,
                     float* __restrict__ gblocks,
                     float2* __restrict__ partials)
{
    __shared__ float ldsx[TILE_ELEMS];
    __shared__ float sblk[TILE_J];

    const int tile = blockIdx.x;          // 0..78
    const int row  = blockIdx.y;          // 0..255
    const int t    = threadIdx.x;

    const long base      = (long)row * IN_DIM + (long)tile * TILE_ELEMS;
    const int  valid     = min(TILE_ELEMS, IN_DIM - tile * TILE_ELEMS);
    const int  validvec  = valid >> 2;    // whole float4s (IN_DIM % 4 == 0)

    // coalesced 128-bit loads of the x tile into LDS (5 float4s per thread)
    for (int i = t; i < TILE_ELEMS / 4; i += 256) {
        float4 v = make_float4(0.f, 0.f, 0.f, 0.f);
        if (i < validvec) v = *(const float4*)(x + base + 4 * i);
        *(float4*)(ldsx + 4 * i) = v;
    }
    __syncthreads();

    // 4 lanes per output block, 20 MACs each
    const int jl = t >> 2;                // local block 0..63
    const int s  = t & 3;                 // sub-slice 0..3
    const int jg = tile * TILE_J + jl;    // global block index

    float val = 0.f;
    if (jg < NBLK) {
        const float* xs = ldsx + jl * BLK_ELEMS + s * 20;
        const float* ss = sv + (long)tile * TILE_ELEMS + jl * BLK_ELEMS + s * 20;
#pragma unroll
        for (int i = 0; i < 20; ++i) val += xs[i] * ss[i];
    }
    // reduce across the 4 sub-slices (within a wave32)
    val += __shfl_down(val, 2, 4);
    val += __shfl_down(val, 1, 4);

    if (s == 0) {
        if (jg < NBLK) gblocks[(long)row * NBLK + jg] = val;
        sblk[jl] = (jg < NBLK) ? val : 0.f;
    }
    __syncthreads();

    // deterministic per-tile partial stats (no atomics)
    if (t == 0) {
        float sm = 0.f, sq = 0.f;
#pragma unroll 8
        for (int j = 0; j < TILE_J; ++j) { float v = sblk[j]; sm += v; sq += v * v; }
        partials[row * NTILES + tile] = make_float2(sm, sq);
    }
}

// ---------------------------------------------------------------------------
// Kernel 2: LayerNorm over the 5000 blocks of each row, in place.
// Grid: (BATCH), Block: 256 threads.
// ---------------------------------------------------------------------------
__global__ __launch_bounds__(256)
void k_layernorm(float* __restrict__ g,
                 const float2* __restrict__ partials,
                 const float* __restrict__ gamma,
                 const float* __restrict__ beta)
{
    __shared__ float s0[256];
    __shared__ float s1[256];
    const int row = blockIdx.x;
    const int t   = threadIdx.x;

    float a = 0.f, b = 0.f;
    for (int i = t; i < NTILES; i += 256) {
        float2 p = partials[row * NTILES + i];
        a += p.x; b += p.y;
    }
    s0[t] = a; s1[t] = b;
    __syncthreads();
    for (int off = 128; off > 0; off >>= 1) {
        if (t < off) { s0[t] += s0[t + off]; s1[t] += s1[t + off]; }
        __syncthreads();
    }
    const float mu   = s0[0] * (1.f / NBLK);
    const float var  = s1[0] * (1.f / NBLK) - mu * mu;
    const float rstd = rsqrtf(var + 1e-5f);

    for (int j = t; j < NBLK; j += 256) {
        const long idx = (long)row * NBLK + j;
        g[idx] = (g[idx] - mu) * rstd * gamma[j] + beta[j];
    }
}

// ---------------------------------------------------------------------------
// Kernel 3: C[M,N] = A[M,K] @ W[N,K]^T + bias  (optional sigmoid)
// fp32 WMMA 16x16x4. Wave computes a 16x16 D tile; 8 waves -> 32x64 per WG.
// A lane fragment: float2 at A[m0 + (lane&15)][k + 2*(lane>>4)]
// B lane fragment: float2 at W[n0 + (lane&15)][k + 2*(lane>>4)]   (row-major W)
// D: VGPR r -> M = m0 + r + 8*(lane>>4), N = n0 + (lane&15)
// Grid: (N/64, M/32), Block: 256. M,N multiples of 32/64; K multiple of 4.
// ---------------------------------------------------------------------------
__global__ __launch_bounds__(256)
void k_wmma_gemm(const float* __restrict__ A,
                 const float* __restrict__ W,
                 const float* __restrict__ bias,
                 float* __restrict__ C,
                 int K, int N, int do_sigmoid)
{
    const int lane  = threadIdx.x & 31;
    const int wave  = threadIdx.x >> 5;
    const int wm    = wave >> 2;            // 0..1
    const int wn    = wave & 3;             // 0..3
    const int m0    = blockIdx.y * 32 + wm * 16;
    const int n0    = blockIdx.x * 64 + wn * 16;
    const int l15   = lane & 15;
    const int khalf = (lane >> 4) * 2;

    const float* ap = A + (long)(m0 + l15) * K + khalf;
    const float* bp = W + (long)(n0 + l15) * K + khalf;

    v8f acc = {};
#pragma unroll 4
    for (int k = 0; k < K; k += 4) {
        v2f a = *(const v2f*)(ap + k);   // 8B-aligned global_load_b64
        v2f b = *(const v2f*)(bp + k);
        acc = __builtin_amdgcn_wmma_f32_16x16x4_f32(
            /*neg_a=*/false, a, /*neg_b=*/false, b,
            /*c_mod=*/(short)0, acc, /*reuse_a=*/false, /*reuse_b=*/false);
    }

    const int   cn = n0 + l15;
    const float bv = bias[cn];
#pragma unroll
    for (int r = 0; r < 8; ++r) {
        const int cm = m0 + r + 8 * (lane >> 4);
        float v = acc[r] + bv;
        if (do_sigmoid) v = 1.f / (1.f + __expf(-v));
        C[(long)cm * N + cn] = v;
    }
}

// ---------------------------------------------------------------------------
// Kernel 4: out[b] = h2[b,:] . W3 + b3   (256 -> 1)
// ---------------------------------------------------------------------------
__global__ __launch_bounds__(256)
void k_final(const float* __restrict__ h2,
             const float* __restrict__ W3,
             const float* __restrict__ b3,
             float* __restrict__ out)
{
    const int b = threadIdx.x;
    float acc = 0.f;
#pragma unroll 8
    for (int k = 0; k < D2; ++k) acc += h2[b * D2 + k] * W3[k];
    out[b] = acc + b3[0];
}

// ---------------------------------------------------------------------------
extern "C" void kernel_launch(void* const* d_in, const int* in_sizes, int n_in,
                              void* d_out, int out_size, void* d_ws, size_t ws_size,
                              hipStream_t stream)
{
    const float* x     = (const float*)d_in[0];
    const float* sv    = (const float*)d_in[1];
    const float* gamma = (const float*)d_in[2];
    const float* beta  = (const float*)d_in[3];
    const float* W1    = (const float*)d_in[4];
    const float* b1    = (const float*)d_in[5];
    const float* W2    = (const float*)d_in[6];
    const float* b2    = (const float*)d_in[7];
    const float* W3    = (const float*)d_in[8];
    const float* b3    = (const float*)d_in[9];
    // d_in[10], d_in[11] (idx0, idx1): structure is known statically.

    // workspace carve-up (~6.6 MB)
    float*  gblocks  = (float*)d_ws;                           // 256*5000
    float2* partials = (float2*)(gblocks + (long)BATCH * NBLK); // 256*79 (8B-aligned)
    float*  h1       = (float*)(partials + BATCH * NTILES);     // 256*1024
    float*  h2       = h1 + (long)BATCH * D1;                   // 256*256

    k_sparse_blocks<<<dim3(NTILES, BATCH), 256, 0, stream>>>(x, sv, gblocks, partials);
    k_layernorm<<<dim3(BATCH), 256, 0, stream>>>(gblocks, partials, gamma, beta);
    k_wmma_gemm<<<dim3(D1 / 64, BATCH / 32), 256, 0, stream>>>(gblocks, W1, b1, h1, NBLK, D1, 0);
    k_wmma_gemm<<<dim3(D2 / 64, BATCH / 32), 256, 0, stream>>>(h1, W2, b2, h2, D1, D2, 1);
    k_final<<<1, 256, 0, stream>>>(h2, W3, b3, (float*)d_out);
}